// FrequencyAwareCrossAttention_3951369912985
// MI455X (gfx1250) — compile-verified
//
#include <hip/hip_runtime.h>
#include <hip/hip_bf16.h>
#include <math.h>

typedef __attribute__((ext_vector_type(2))) float v2f;
typedef __attribute__((ext_vector_type(8))) float v8f;

// f32 WMMA: D(16x16) = A(16x4) * B(4x16) + C   (full-precision matrix pipe)
#define WMMA_F32_K4(a, b, c) \
  __builtin_amdgcn_wmma_f32_16x16x4_f32(false, (a), false, (b), (short)0, (c), false, false)

constexpr int Bb = 8, T = 128, TK = 256, J = 17, D = 256, NH = 8, C = 32;
constexpr int NG = Bb * NH * J;        // 1088 attention groups
constexpr int KEEPQ = 32, KEEPKV = 64, FQ_FULL = 65;
constexpr float SCALE = 0.17677669529663687f;   // C^-0.5

__device__ inline float sigmoidf_(float x) { return 1.0f / (1.0f + expf(-x)); }

// ---------------------------------------------------------------- DFT tables
__global__ void init_dft(float* fq_cos, float* fq_sin, float* fk_cos, float* fk_sin,
                         float* if_cos, float* if_sin) {
  int i = blockIdx.x * blockDim.x + threadIdx.x;
  const float PI2 = 6.283185307179586f;
  if (i < KEEPQ * T) {                       // forward DFT, q side (32 x 128)
    int f = i / T, tt = i % T;
    float th = PI2 * (float)f * (float)tt / (float)T;
    fq_cos[i] = cosf(th);
    fq_sin[i] = -sinf(th);
  }
  if (i < KEEPKV * TK) {                     // forward DFT, kv side (64 x 256)
    int f = i / TK, tt = i % TK;
    float th = PI2 * (float)f * (float)tt / (float)TK;
    fk_cos[i] = cosf(th);
    fk_sin[i] = -sinf(th);
  }
  if (i < T * KEEPQ) {                       // inverse rDFT (128 x 32), hermitian weights
    int tt = i / KEEPQ, f = i % KEEPQ;
    float w = (f == 0) ? 1.0f : 2.0f;
    float th = PI2 * (float)f * (float)tt / (float)T;
    if_cos[i] = w * cosf(th) / (float)T;
    if_sin[i] = -w * sinf(th) / (float)T;
  }
}

// ------------------------------------------- projection GEMM -> head layout
// X: (n_tokens x 256) row-major, W: 256x256 row-major.
// Out[b,h,j,tt,c] = (X @ W)[token, h*32+c]
// TDIM is compile-time so the token->(b,tt,jj) divisions strength-reduce to
// magic-number multiplies (no v_s_rcp_f32 integer-division chains).
template <int TDIM>
__global__ void proj_heads(const float* __restrict__ X, const float* __restrict__ W,
                           float* __restrict__ Out) {
  int tm = blockIdx.x, tn = blockIdx.y;
  int lane = threadIdx.x & 31, half = lane >> 4, mrow = lane & 15;
  int row = tm * 16 + mrow;
  int col = tn * 16 + mrow;
  const float* arow = X + (size_t)row * D;
  v8f acc = {};
  for (int k0 = 0; k0 < D; k0 += 4) {
    int k = k0 + 2 * half;
    v2f a = {arow[k], arow[k + 1]};
    v2f b = {W[(size_t)k * D + col], W[(size_t)(k + 1) * D + col]};
    acc = WMMA_F32_K4(a, b, acc);
  }
  int mcol = tn * 16 + mrow;
  int h = mcol >> 5, c = mcol & 31;
  for (int r = 0; r < 8; ++r) {
    int m = tm * 16 + r + 8 * half;          // token index
    int b_i = m / (TDIM * J);                // constant divisor -> mul_hi
    int rem = m % (TDIM * J);
    int tt = rem / J, jj = rem % J;
    Out[(((size_t)(b_i * NH + h) * J + jj) * TDIM + tt) * C + c] = acc[r];
  }
}

// -------------------------- fused time attention: QK^T -> softmax -> PV
// Also writes (1-gate)*P into attn output (resize kernel accumulates on top).
__global__ void attn_time(const float* __restrict__ QP, const float* __restrict__ KP,
                          const float* __restrict__ VP, const float* __restrict__ gatep,
                          float* __restrict__ OUT_TIME, float* __restrict__ attn_out) {
  extern __shared__ float sS[];              // T*TK = 128KB (of 320KB WGP LDS)
  int g = blockIdx.x;
  const float* qg = QP + (size_t)g * T * C;
  const float* kg = KP + (size_t)g * TK * C;
  const float* vg = VP + (size_t)g * TK * C;
  int wave = threadIdx.x >> 5;
  int lane = threadIdx.x & 31, half = lane >> 4, mrow = lane & 15;
  int m0 = wave * 16;                        // 8 waves x 16 rows = 128 q rows
  float gate = sigmoidf_(gatep[0]);

  // S = (Q K^T) * scale
  for (int tn = 0; tn < TK / 16; ++tn) {
    v8f acc = {};
    const float* ar = qg + (size_t)(m0 + mrow) * C;
    const float* br = kg + (size_t)(tn * 16 + mrow) * C;
    for (int k0 = 0; k0 < C; k0 += 4) {
      int k = k0 + 2 * half;
      v2f a = {ar[k], ar[k + 1]};
      v2f b = {br[k], br[k + 1]};
      acc = WMMA_F32_K4(a, b, acc);
    }
    for (int r = 0; r < 8; ++r)
      sS[(size_t)(m0 + r + 8 * half) * TK + tn * 16 + mrow] = acc[r] * SCALE;
  }
  __syncthreads();

  // row softmax; emit (1-gate)*P into attn output
  if (threadIdx.x < T) {
    float* rowp = sS + (size_t)threadIdx.x * TK;
    float mx = -3.4e38f;
    for (int k = 0; k < TK; ++k) mx = fmaxf(mx, rowp[k]);
    float s = 0.f;
    for (int k = 0; k < TK; ++k) { float e = expf(rowp[k] - mx); rowp[k] = e; s += e; }
    float inv = 1.0f / s;
    float* ao = attn_out + (size_t)g * T * TK + (size_t)threadIdx.x * TK;
    for (int k = 0; k < TK; ++k) { float p = rowp[k] * inv; rowp[k] = p; ao[k] = (1.0f - gate) * p; }
  }
  __syncthreads();

  // O = P V
  for (int tn = 0; tn < C / 16; ++tn) {
    v8f acc = {};
    const float* pr = sS + (size_t)(m0 + mrow) * TK;
    for (int k0 = 0; k0 < TK; k0 += 4) {
      int k = k0 + 2 * half;
      v2f a = {pr[k], pr[k + 1]};
      v2f b = {vg[(size_t)k * C + tn * 16 + mrow], vg[(size_t)(k + 1) * C + tn * 16 + mrow]};
      acc = WMMA_F32_K4(a, b, acc);
    }
    for (int r = 0; r < 8; ++r)
      OUT_TIME[(size_t)g * T * C + (size_t)(m0 + r + 8 * half) * C + tn * 16 + mrow] = acc[r];
  }
}

// ------------------------------- rFFT(q) as DFT-matrix GEMM: (32x128)x(128x32)
__global__ void rfft_q(const float* __restrict__ QP, const float* __restrict__ FQC,
                       const float* __restrict__ FQS, float* __restrict__ QFR,
                       float* __restrict__ QFI) {
  int g = blockIdx.x;
  const float* xg = QP + (size_t)g * T * C;
  int lane = threadIdx.x & 31, half = lane >> 4, mrow = lane & 15;
  for (int mt = 0; mt < KEEPQ / 16; ++mt)
    for (int nt = 0; nt < C / 16; ++nt) {
      v8f accR = {}, accI = {};
      const float* crow = FQC + (size_t)(mt * 16 + mrow) * T;
      const float* srow = FQS + (size_t)(mt * 16 + mrow) * T;
      for (int k0 = 0; k0 < T; k0 += 4) {
        int k = k0 + 2 * half;
        v2f ac = {crow[k], crow[k + 1]};
        v2f as = {srow[k], srow[k + 1]};
        v2f b = {xg[(size_t)k * C + nt * 16 + mrow], xg[(size_t)(k + 1) * C + nt * 16 + mrow]};
        accR = WMMA_F32_K4(ac, b, accR);
        accI = WMMA_F32_K4(as, b, accI);
      }
      for (int r = 0; r < 8; ++r) {
        size_t off = (size_t)g * KEEPQ * C + (size_t)(mt * 16 + r + 8 * half) * C + nt * 16 + mrow;
        QFR[off] = accR[r];
        QFI[off] = accI[r];
      }
    }
}

// ----------------------------- rFFT(k) and rFFT(v): (64x256)x(256x32) each
__global__ void rfft_kv(const float* __restrict__ KP, const float* __restrict__ VP,
                        const float* __restrict__ FKC, const float* __restrict__ FKS,
                        float* __restrict__ KFR, float* __restrict__ KFI,
                        float* __restrict__ VFR, float* __restrict__ VFI) {
  int g = blockIdx.x;
  const float* xg = (blockIdx.y == 0 ? KP : VP) + (size_t)g * TK * C;
  float* outR = (blockIdx.y == 0 ? KFR : VFR);
  float* outI = (blockIdx.y == 0 ? KFI : VFI);
  int lane = threadIdx.x & 31, half = lane >> 4, mrow = lane & 15;
  for (int mt = 0; mt < KEEPKV / 16; ++mt)
    for (int nt = 0; nt < C / 16; ++nt) {
      v8f accR = {}, accI = {};
      const float* crow = FKC + (size_t)(mt * 16 + mrow) * TK;
      const float* srow = FKS + (size_t)(mt * 16 + mrow) * TK;
      for (int k0 = 0; k0 < TK; k0 += 4) {
        int k = k0 + 2 * half;
        v2f ac = {crow[k], crow[k + 1]};
        v2f as = {srow[k], srow[k + 1]};
        v2f b = {xg[(size_t)k * C + nt * 16 + mrow], xg[(size_t)(k + 1) * C + nt * 16 + mrow]};
        accR = WMMA_F32_K4(ac, b, accR);
        accI = WMMA_F32_K4(as, b, accI);
      }
      for (int r = 0; r < 8; ++r) {
        size_t off = (size_t)g * KEEPKV * C + (size_t)(mt * 16 + r + 8 * half) * C + nt * 16 + mrow;
        outR[off] = accR[r];
        outI[off] = accI[r];
      }
    }
}

// ---------- complex frequency attention: af = qf conj(kf)*scale; split softmax; of = af vf
__global__ void attn_freq(const float* __restrict__ QFR, const float* __restrict__ QFI,
                          const float* __restrict__ KFR, const float* __restrict__ KFI,
                          const float* __restrict__ VFR, const float* __restrict__ VFI,
                          float* __restrict__ AFR, float* __restrict__ OFR,
                          float* __restrict__ OFI) {
  __shared__ float ar[KEEPQ * KEEPKV];       // 8KB
  __shared__ float ai[KEEPQ * KEEPKV];       // 8KB
  int g = blockIdx.x;
  int lane = threadIdx.x & 31, half = lane >> 4, mrow = lane & 15;
  const float* qr = QFR + (size_t)g * KEEPQ * C;
  const float* qi = QFI + (size_t)g * KEEPQ * C;
  const float* kr = KFR + (size_t)g * KEEPKV * C;
  const float* ki = KFI + (size_t)g * KEEPKV * C;
  const float* vr = VFR + (size_t)g * KEEPKV * C;
  const float* vi = VFI + (size_t)g * KEEPKV * C;

  // af = qf * conj(kf): re = qr.kr + qi.ki ; im = qi.kr - qr.ki
  for (int mt = 0; mt < KEEPQ / 16; ++mt)
    for (int nt = 0; nt < KEEPKV / 16; ++nt) {
      v8f accR = {}, accI = {};
      const float* qrr = qr + (size_t)(mt * 16 + mrow) * C;
      const float* qir = qi + (size_t)(mt * 16 + mrow) * C;
      const float* krr = kr + (size_t)(nt * 16 + mrow) * C;
      const float* kir = ki + (size_t)(nt * 16 + mrow) * C;
      for (int k0 = 0; k0 < C; k0 += 4) {
        int k = k0 + 2 * half;
        v2f aR = {qrr[k], qrr[k + 1]};
        v2f aI = {qir[k], qir[k + 1]};
        v2f aRn = {-aR.x, -aR.y};            // f32 WMMA has no A-neg modifier; do it in VALU
        v2f bR = {krr[k], krr[k + 1]};
        v2f bI = {kir[k], kir[k + 1]};
        accR = WMMA_F32_K4(aR, bR, accR);
        accR = WMMA_F32_K4(aI, bI, accR);
        accI = WMMA_F32_K4(aI, bR, accI);
        accI = WMMA_F32_K4(aRn, bI, accI);
      }
      for (int r = 0; r < 8; ++r) {
        int o = (mt * 16 + r + 8 * half) * KEEPKV + nt * 16 + mrow;
        ar[o] = accR[r] * SCALE;
        ai[o] = accI[r] * SCALE;
      }
    }
  __syncthreads();

  // separate softmax on real & imag; store softmaxed real part for the resize
  {
    int row = lane;                          // 32 rows, 32 lanes
    float* rr = ar + row * KEEPKV;
    float* ri = ai + row * KEEPKV;
    float mR = -3.4e38f, mI = -3.4e38f;
    for (int k = 0; k < KEEPKV; ++k) { mR = fmaxf(mR, rr[k]); mI = fmaxf(mI, ri[k]); }
    float sR = 0.f, sI = 0.f;
    for (int k = 0; k < KEEPKV; ++k) {
      float eR = expf(rr[k] - mR), eI = expf(ri[k] - mI);
      rr[k] = eR; ri[k] = eI; sR += eR; sI += eI;
    }
    float iR = 1.0f / sR, iI = 1.0f / sI;
    float* afo = AFR + (size_t)g * KEEPQ * KEEPKV + (size_t)row * KEEPKV;
    for (int k = 0; k < KEEPKV; ++k) { rr[k] *= iR; ri[k] *= iI; afo[k] = rr[k]; }
  }
  __syncthreads();

  // of = af * vf: re = ar.vr - ai.vi ; im = ar.vi + ai.vr
  for (int mt = 0; mt < KEEPQ / 16; ++mt)
    for (int nt = 0; nt < C / 16; ++nt) {
      v8f accR = {}, accI = {};
      const float* arr = ar + (mt * 16 + mrow) * KEEPKV;
      const float* air = ai + (mt * 16 + mrow) * KEEPKV;
      for (int k0 = 0; k0 < KEEPKV; k0 += 4) {
        int k = k0 + 2 * half;
        v2f aR = {arr[k], arr[k + 1]};
        v2f aI = {air[k], air[k + 1]};
        v2f aIn = {-aI.x, -aI.y};
        v2f bR = {vr[(size_t)k * C + nt * 16 + mrow], vr[(size_t)(k + 1) * C + nt * 16 + mrow]};
        v2f bI = {vi[(size_t)k * C + nt * 16 + mrow], vi[(size_t)(k + 1) * C + nt * 16 + mrow]};
        accR = WMMA_F32_K4(aR, bR, accR);
        accR = WMMA_F32_K4(aIn, bI, accR);
        accI = WMMA_F32_K4(aR, bI, accI);
        accI = WMMA_F32_K4(aI, bR, accI);
      }
      for (int r = 0; r < 8; ++r) {
        size_t o = (size_t)g * KEEPQ * C + (size_t)(mt * 16 + r + 8 * half) * C + nt * 16 + mrow;
        OFR[o] = accR[r];
        OFI[o] = accI[r];
      }
    }
}

// -------- inverse-rDFT GEMM + gated blend with out_time -> (b,t,j,d) layout
__global__ void irfft_combine(const float* __restrict__ OFR, const float* __restrict__ OFI,
                              const float* __restrict__ IFC, const float* __restrict__ IFS,
                              const float* __restrict__ OUT_TIME, const float* __restrict__ gatep,
                              float* __restrict__ OUT_COMB) {
  int g = blockIdx.x;
  int lane = threadIdx.x & 31, half = lane >> 4, mrow = lane & 15;
  int jj = g % J, bh = g / J, h = bh % NH, b_i = bh / NH;
  float gate = sigmoidf_(gatep[0]);
  const float* ofr = OFR + (size_t)g * KEEPQ * C;
  const float* ofi = OFI + (size_t)g * KEEPQ * C;
  for (int mt = 0; mt < T / 16; ++mt)
    for (int nt = 0; nt < C / 16; ++nt) {
      v8f acc = {};
      const float* crow = IFC + (size_t)(mt * 16 + mrow) * KEEPQ;
      const float* srow = IFS + (size_t)(mt * 16 + mrow) * KEEPQ;
      for (int k0 = 0; k0 < KEEPQ; k0 += 4) {
        int k = k0 + 2 * half;
        v2f ac = {crow[k], crow[k + 1]};
        v2f as = {srow[k], srow[k + 1]};
        v2f bR = {ofr[(size_t)k * C + nt * 16 + mrow], ofr[(size_t)(k + 1) * C + nt * 16 + mrow]};
        v2f bI = {ofi[(size_t)k * C + nt * 16 + mrow], ofi[(size_t)(k + 1) * C + nt * 16 + mrow]};
        acc = WMMA_F32_K4(ac, bR, acc);
        acc = WMMA_F32_K4(as, bI, acc);
      }
      for (int r = 0; r < 8; ++r) {
        int tt = mt * 16 + r + 8 * half, cc = nt * 16 + mrow;
        float tv = OUT_TIME[(size_t)g * T * C + (size_t)tt * C + cc];
        float comb = gate * acc[r] + (1.0f - gate) * tv;
        OUT_COMB[(((size_t)b_i * T + tt) * J + jj) * D + h * C + cc] = comb;
      }
    }
}

// ------- bilinear (half-pixel) resize of padded af.real, accumulate into attn
__global__ void resize_add(const float* __restrict__ AFR, const float* __restrict__ gatep,
                           float* __restrict__ attn_out) {
  size_t idx = (size_t)blockIdx.x * blockDim.x + threadIdx.x;
  size_t total = (size_t)NG * T * TK;
  if (idx >= total) return;
  int k = (int)(idx % TK);
  int tt = (int)((idx / TK) % T);
  int g = (int)(idx / ((size_t)T * TK));
  float gate = sigmoidf_(gatep[0]);
  float sy = (tt + 0.5f) * ((float)FQ_FULL / (float)T) - 0.5f;
  float sx = (k + 0.5f) * ((float)KEEPKV / (float)TK) - 0.5f;
  int y0 = (int)floorf(sy); float fy = sy - (float)y0;
  int x0 = (int)floorf(sx); float fx = sx - (float)x0;
  int y1 = y0 + 1, x1 = x0 + 1;
  y0 = min(max(y0, 0), FQ_FULL - 1); y1 = min(max(y1, 0), FQ_FULL - 1);
  x0 = min(max(x0, 0), KEEPKV - 1);  x1 = min(max(x1, 0), KEEPKV - 1);
  const float* a = AFR + (size_t)g * KEEPQ * KEEPKV;
  auto fetch = [&](int y, int x) -> float { return (y < KEEPQ) ? a[y * KEEPKV + x] : 0.0f; };
  float v00 = fetch(y0, x0), v01 = fetch(y0, x1), v10 = fetch(y1, x0), v11 = fetch(y1, x1);
  float v = (1.f - fy) * ((1.f - fx) * v00 + fx * v01) + fy * ((1.f - fx) * v10 + fx * v11);
  attn_out[idx] += gate * v;
}

// ---------------------------------------------- final projection + bias
__global__ void final_proj(const float* __restrict__ X, const float* __restrict__ W,
                           const float* __restrict__ bias, float* __restrict__ Out) {
  int tm = blockIdx.x, tn = blockIdx.y;
  int lane = threadIdx.x & 31, half = lane >> 4, mrow = lane & 15;
  int row = tm * 16 + mrow, col = tn * 16 + mrow;
  const float* arow = X + (size_t)row * D;
  v8f acc = {};
  for (int k0 = 0; k0 < D; k0 += 4) {
    int k = k0 + 2 * half;
    v2f a = {arow[k], arow[k + 1]};
    v2f b = {W[(size_t)k * D + col], W[(size_t)(k + 1) * D + col]};
    acc = WMMA_F32_K4(a, b, acc);
  }
  for (int r = 0; r < 8; ++r) {
    int m = tm * 16 + r + 8 * half;
    int n = tn * 16 + mrow;
    Out[(size_t)m * D + n] = acc[r] + bias[n];
  }
}

extern "C" void kernel_launch(void* const* d_in, const int* in_sizes, int n_in,
                              void* d_out, int out_size, void* d_ws, size_t ws_size,
                              hipStream_t stream) {
  (void)in_sizes; (void)n_in; (void)out_size; (void)ws_size;
  const float* q        = (const float*)d_in[0];
  const float* kv       = (const float*)d_in[1];
  const float* wq       = (const float*)d_in[2];
  const float* wk       = (const float*)d_in[3];
  const float* wv       = (const float*)d_in[4];
  const float* w_proj   = (const float*)d_in[5];
  const float* b_proj   = (const float*)d_in[6];
  const float* freq_gate= (const float*)d_in[7];

  float* ws = (float*)d_ws;
  size_t off = 0;
  auto alloc = [&](size_t n) { float* p = ws + off; off += n; return p; };
  float* QP       = alloc((size_t)NG * T * C);
  float* KP       = alloc((size_t)NG * TK * C);
  float* VP       = alloc((size_t)NG * TK * C);
  float* OUT_TIME = alloc((size_t)NG * T * C);
  float* QFR      = alloc((size_t)NG * KEEPQ * C);
  float* QFI      = alloc((size_t)NG * KEEPQ * C);
  float* KFR      = alloc((size_t)NG * KEEPKV * C);
  float* KFI      = alloc((size_t)NG * KEEPKV * C);
  float* VFR      = alloc((size_t)NG * KEEPKV * C);
  float* VFI      = alloc((size_t)NG * KEEPKV * C);
  float* AFR      = alloc((size_t)NG * KEEPQ * KEEPKV);
  float* OFR      = alloc((size_t)NG * KEEPQ * C);
  float* OFI      = alloc((size_t)NG * KEEPQ * C);
  float* OUT_COMB = alloc((size_t)Bb * T * J * D);
  float* FQC = alloc(KEEPQ * T);   float* FQS = alloc(KEEPQ * T);
  float* FKC = alloc(KEEPKV * TK); float* FKS = alloc(KEEPKV * TK);
  float* IFC = alloc(T * KEEPQ);   float* IFS = alloc(T * KEEPQ);

  float* attn_out = (float*)d_out;
  float* out_out  = (float*)d_out + (size_t)NG * T * TK;

  init_dft<<<64, 256, 0, stream>>>(FQC, FQS, FKC, FKS, IFC, IFS);

  dim3 gq(Bb * T * J / 16, D / 16);
  proj_heads<T><<<gq, 32, 0, stream>>>(q, wq, QP);
  dim3 gk(Bb * TK * J / 16, D / 16);
  proj_heads<TK><<<gk, 32, 0, stream>>>(kv, wk, KP);
  proj_heads<TK><<<gk, 32, 0, stream>>>(kv, wv, VP);

  attn_time<<<NG, 256, T * TK * sizeof(float), stream>>>(QP, KP, VP, freq_gate,
                                                         OUT_TIME, attn_out);

  rfft_q<<<NG, 32, 0, stream>>>(QP, FQC, FQS, QFR, QFI);
  dim3 gkv(NG, 2);
  rfft_kv<<<gkv, 32, 0, stream>>>(KP, VP, FKC, FKS, KFR, KFI, VFR, VFI);

  attn_freq<<<NG, 32, 0, stream>>>(QFR, QFI, KFR, KFI, VFR, VFI, AFR, OFR, OFI);

  irfft_combine<<<NG, 32, 0, stream>>>(OFR, OFI, IFC, IFS, OUT_TIME, freq_gate, OUT_COMB);

  size_t total_attn = (size_t)NG * T * TK;
  resize_add<<<(unsigned)((total_attn + 255) / 256), 256, 0, stream>>>(AFR, freq_gate, attn_out);

  dim3 gp(Bb * T * J / 16, D / 16);
  final_proj<<<gp, 32, 0, stream>>>(OUT_COMB, w_proj, b_proj, out_out);
}